// CoAttention_41266045780440
// MI455X (gfx1250) — compile-verified
//
#include <hip/hip_runtime.h>
#include <math.h>

typedef __attribute__((ext_vector_type(2))) float v2f;
typedef __attribute__((ext_vector_type(8))) float v8f;

#define B_   16
#define LV_  196
#define LQ_  32
#define DIM_ 512

__device__ __forceinline__ float dev_tanh(float x) {
#if __has_builtin(__builtin_amdgcn_tanhf)
  return __builtin_amdgcn_tanhf(x);           // gfx1250 v_tanh_f32
#elif __has_builtin(__builtin_amdgcn_tanh_f32)
  return __builtin_amdgcn_tanh_f32(x);
#else
  return tanhf(x);
#endif
}

// ---------------------------------------------------------------------------
// Y[M,512] = X[M,512] @ W[512,512]^T + bias   (nn.Linear), fp32 WMMA 16x16x4.
// One wave computes a 16(M) x 64(N) strip: 4 accumulator tiles, K-loop of 4.
// tasks = (M/16) * 8 n-strips; 8 waves / block -> grid = M/16 blocks.
// ---------------------------------------------------------------------------
__global__ __launch_bounds__(256) void proj_wmma(const float* __restrict__ X,
                                                 const float* __restrict__ W,
                                                 const float* __restrict__ bias,
                                                 float* __restrict__ Y, int M) {
  const int K = DIM_, N = DIM_;
  const int lane = threadIdx.x & 31;
  const int wave = threadIdx.x >> 5;
  const int task = blockIdx.x * 8 + wave;
  const int m0 = (task >> 3) * 16;
  const int n0 = (task & 7) * 64;
  if (m0 >= M) return;                       // wave-uniform
  const int mrow = lane & 15;
  const int kh   = (lane >> 4) << 1;         // 0 or 2
  const int ncol = lane & 15;

  v8f acc[4] = {};
  const float* xr = X + (size_t)(m0 + mrow) * K + kh;
  const float* w0 = W + (size_t)(n0 +  0 + ncol) * K + kh;
  const float* w1 = W + (size_t)(n0 + 16 + ncol) * K + kh;
  const float* w2 = W + (size_t)(n0 + 32 + ncol) * K + kh;
  const float* w3 = W + (size_t)(n0 + 48 + ncol) * K + kh;

  for (int k = 0; k < K; k += 4) {
    v2f a  = *(const v2f*)(xr + k);
    v2f b0 = *(const v2f*)(w0 + k);
    v2f b1 = *(const v2f*)(w1 + k);
    v2f b2 = *(const v2f*)(w2 + k);
    v2f b3 = *(const v2f*)(w3 + k);
    acc[0] = __builtin_amdgcn_wmma_f32_16x16x4_f32(false, a, false, b0, (short)0, acc[0], false, false);
    acc[1] = __builtin_amdgcn_wmma_f32_16x16x4_f32(false, a, false, b1, (short)0, acc[1], false, false);
    acc[2] = __builtin_amdgcn_wmma_f32_16x16x4_f32(false, a, false, b2, (short)0, acc[2], false, false);
    acc[3] = __builtin_amdgcn_wmma_f32_16x16x4_f32(false, a, false, b3, (short)0, acc[3], false, false);
  }

  const int hi = lane >> 4;
#pragma unroll
  for (int s = 0; s < 4; ++s) {
    float bval = bias[n0 + s * 16 + ncol];
#pragma unroll
    for (int r = 0; r < 8; ++r) {
      Y[(size_t)(m0 + r + 8 * hi) * N + n0 + s * 16 + ncol] = acc[s][r] + bval;
    }
  }
}

// ---------------------------------------------------------------------------
// Additive-attention scores. One block per (b,q); q-row + head weights in LDS.
// Wave w handles v = w, w+8, ... ; per-v wave32 shuffle reduction over d.
// sv[b,q,v] = Whv . tanh(v2[b,v,:]+q2[b,q,:]) + bhv ; same for sq.
// ---------------------------------------------------------------------------
__global__ __launch_bounds__(256) void score_kernel(const float* __restrict__ v2,
                                                    const float* __restrict__ q2,
                                                    const float* __restrict__ Whv,
                                                    const float* __restrict__ Whq,
                                                    const float* __restrict__ bhv,
                                                    const float* __restrict__ bhq,
                                                    float* __restrict__ sv,
                                                    float* __restrict__ sq) {
  __shared__ float qs[DIM_], wv[DIM_], wq[DIM_];
  const int b = blockIdx.x / LQ_;
  const int q = blockIdx.x % LQ_;
  for (int d = threadIdx.x; d < DIM_; d += 256) {
    qs[d] = q2[((size_t)b * LQ_ + q) * DIM_ + d];
    wv[d] = Whv[d];
    wq[d] = Whq[d];
  }
  __syncthreads();
  const int wave = threadIdx.x >> 5;
  const int lane = threadIdx.x & 31;
  const float cv = bhv[0], cq = bhq[0];
  for (int v = wave; v < LV_; v += 8) {
    const float* vr = v2 + ((size_t)b * LV_ + v) * DIM_;
    float s1 = 0.f, s2 = 0.f;
    for (int d = lane; d < DIM_; d += 32) {
      float t = dev_tanh(vr[d] + qs[d]);
      s1 += wv[d] * t;
      s2 += wq[d] * t;
    }
    for (int off = 16; off; off >>= 1) {
      s1 += __shfl_down(s1, off, 32);
      s2 += __shfl_down(s2, off, 32);
    }
    if (lane == 0) {
      size_t idx = ((size_t)b * LQ_ + q) * LV_ + v;
      sv[idx] = s1 + cv;
      sq[idx] = s2 + cq;
    }
  }
}

// softmax over last axis (length L=196) of s[nrows, L], in place; wave per row.
__global__ __launch_bounds__(256) void softmax_rows(float* __restrict__ s,
                                                    int nrows, int L) {
  const int wave = threadIdx.x >> 5, lane = threadIdx.x & 31;
  const int row = blockIdx.x * 8 + wave;
  if (row >= nrows) return;
  float* p = s + (size_t)row * L;
  float mx = -__builtin_inff();
  for (int i = lane; i < L; i += 32) mx = fmaxf(mx, p[i]);
  for (int off = 16; off; off >>= 1) mx = fmaxf(mx, __shfl_xor(mx, off, 32));
  float e[7];
  float sum = 0.f;
  int c = 0;
  for (int i = lane; i < L; i += 32) { float ex = __expf(p[i] - mx); e[c++] = ex; sum += ex; }
  for (int off = 16; off; off >>= 1) sum += __shfl_xor(sum, off, 32);
  float inv = 1.f / sum;
  c = 0;
  for (int i = lane; i < L; i += 32) p[i] = e[c++] * inv;
}

// softmax over q-axis (len 32 == one wave) of sq[B, LQ, LV], in place.
// One wave per (b, v) column; lane == q.
__global__ __launch_bounds__(256) void softmax_cols(float* __restrict__ s) {
  const int wave = threadIdx.x >> 5, lane = threadIdx.x & 31;
  const int idx = blockIdx.x * 8 + wave;
  if (idx >= B_ * LV_) return;
  const int b = idx / LV_, v = idx % LV_;
  float* base = s + (size_t)b * LQ_ * LV_ + v;
  float x = base[(size_t)lane * LV_];
  float mx = x;
  for (int off = 16; off; off >>= 1) mx = fmaxf(mx, __shfl_xor(mx, off, 32));
  float e = __expf(x - mx);
  float sum = e;
  for (int off = 16; off; off >>= 1) sum += __shfl_xor(sum, off, 32);
  base[(size_t)lane * LV_] = e / sum;
}

// ---------------------------------------------------------------------------
// v_att[b] = av[b] (32x196)  @ v2[b] (196x512).  K=196 (49 WMMA steps).
// 16 tasks per b (2 m-tiles x 8 n-strips) -> 256 wave-tasks -> 32 blocks.
// ---------------------------------------------------------------------------
__global__ __launch_bounds__(256) void attn_v(const float* __restrict__ av,
                                              const float* __restrict__ v2,
                                              float* __restrict__ out) {
  const int lane = threadIdx.x & 31;
  const int task = blockIdx.x * 8 + (threadIdx.x >> 5);
  const int b = task >> 4;
  const int t = task & 15;
  const int m0 = (t >> 3) * 16;
  const int n0 = (t & 7) * 64;
  const float* A  = av + (size_t)b * LQ_ * LV_;
  const float* Bm = v2 + (size_t)b * LV_ * DIM_;
  const int mrow = lane & 15, kh = (lane >> 4) << 1, ncol = lane & 15;

  v8f acc[4] = {};
  for (int k = 0; k < LV_; k += 4) {
    v2f a = *(const v2f*)(A + (size_t)(m0 + mrow) * LV_ + k + kh);
#pragma unroll
    for (int s = 0; s < 4; ++s) {
      const int n = n0 + s * 16 + ncol;
      v2f bb;
      bb.x = Bm[(size_t)(k + kh) * DIM_ + n];
      bb.y = Bm[(size_t)(k + kh + 1) * DIM_ + n];
      acc[s] = __builtin_amdgcn_wmma_f32_16x16x4_f32(false, a, false, bb, (short)0, acc[s], false, false);
    }
  }
  const int hi = lane >> 4;
  float* O = out + (size_t)b * LQ_ * DIM_;
#pragma unroll
  for (int s = 0; s < 4; ++s)
#pragma unroll
    for (int r = 0; r < 8; ++r)
      O[(size_t)(m0 + r + 8 * hi) * DIM_ + n0 + s * 16 + ncol] = acc[s][r];
}

// ---------------------------------------------------------------------------
// q_att[b,v,d] = sum_q aq[b,q,v] * q2[b,q,d]:  A = aq[b]^T (196x32), K=32.
// M padded to 208 (13 m-tiles); loads clamped, stores guarded.
// 104 tasks per b -> 1664 wave-tasks -> 208 blocks.
// ---------------------------------------------------------------------------
__global__ __launch_bounds__(256) void attn_q(const float* __restrict__ aq,
                                              const float* __restrict__ q2,
                                              float* __restrict__ out) {
  const int lane = threadIdx.x & 31;
  const int task = blockIdx.x * 8 + (threadIdx.x >> 5);
  const int b = task / 104;
  const int t = task % 104;
  const int m0 = (t >> 3) * 16;      // 0..192
  const int n0 = (t & 7) * 64;
  const float* A  = aq + (size_t)b * LQ_ * LV_;   // A[m][k] = aq[b][k][m]
  const float* Bm = q2 + (size_t)b * LQ_ * DIM_;
  const int mrow = lane & 15, kh = (lane >> 4) << 1, ncol = lane & 15;
  const int m  = m0 + mrow;
  const int mc = m < LV_ ? m : (LV_ - 1);         // clamp pad-row loads

  v8f acc[4] = {};
  for (int k = 0; k < LQ_; k += 4) {
    v2f a;
    a.x = A[(size_t)(k + kh) * LV_ + mc];
    a.y = A[(size_t)(k + kh + 1) * LV_ + mc];
#pragma unroll
    for (int s = 0; s < 4; ++s) {
      const int n = n0 + s * 16 + ncol;
      v2f bb;
      bb.x = Bm[(size_t)(k + kh) * DIM_ + n];
      bb.y = Bm[(size_t)(k + kh + 1) * DIM_ + n];
      acc[s] = __builtin_amdgcn_wmma_f32_16x16x4_f32(false, a, false, bb, (short)0, acc[s], false, false);
    }
  }
  const int hi = lane >> 4;
  float* O = out + (size_t)b * LV_ * DIM_;
#pragma unroll
  for (int s = 0; s < 4; ++s)
#pragma unroll
    for (int r = 0; r < 8; ++r) {
      const int row = m0 + r + 8 * hi;
      if (row < LV_)
        O[(size_t)row * DIM_ + n0 + s * 16 + ncol] = acc[s][r];
    }
}

// ---------------------------------------------------------------------------
extern "C" void kernel_launch(void* const* d_in, const int* in_sizes, int n_in,
                              void* d_out, int out_size, void* d_ws, size_t ws_size,
                              hipStream_t stream) {
  (void)in_sizes; (void)n_in; (void)out_size; (void)ws_size;
  const float* v   = (const float*)d_in[0];
  const float* q   = (const float*)d_in[1];
  const float* Wv  = (const float*)d_in[2];
  const float* bv  = (const float*)d_in[3];
  const float* Wq  = (const float*)d_in[4];
  const float* bq  = (const float*)d_in[5];
  const float* Whv = (const float*)d_in[6];
  const float* bhv = (const float*)d_in[7];
  const float* Whq = (const float*)d_in[8];
  const float* bhq = (const float*)d_in[9];

  float* ws = (float*)d_ws;
  float* v2 = ws;                                        // 16*196*512
  float* q2 = v2 + (size_t)B_ * LV_ * DIM_;              // 16*32*512
  float* sv = q2 + (size_t)B_ * LQ_ * DIM_;              // 16*32*196
  float* sq = sv + (size_t)B_ * LQ_ * LV_;               // 16*32*196

  float* out_v = (float*)d_out;                          // [B,LQ,DIM]
  float* out_q = out_v + (size_t)B_ * LQ_ * DIM_;        // [B,LV,DIM]

  proj_wmma   <<<(B_ * LV_) / 16, 256, 0, stream>>>(v, Wv, bv, v2, B_ * LV_);
  proj_wmma   <<<(B_ * LQ_) / 16, 256, 0, stream>>>(q, Wq, bq, q2, B_ * LQ_);
  score_kernel<<<B_ * LQ_,        256, 0, stream>>>(v2, q2, Whv, Whq, bhv, bhq, sv, sq);
  softmax_rows<<<(B_ * LQ_) / 8,  256, 0, stream>>>(sv, B_ * LQ_, LV_);
  softmax_cols<<<(B_ * LV_ + 7) / 8, 256, 0, stream>>>(sq);
  attn_v      <<<32,  256, 0, stream>>>(sv, v2, out_v);
  attn_q      <<<208, 256, 0, stream>>>(sq, q2, out_q);
}